// QuantumLayer_34385508172522
// MI455X (gfx1250) — compile-verified
//
#include <hip/hip_runtime.h>

// 10-qubit statevector simulator, one batch element per wave32.
// State: 1024 complex amplitudes = 64 f32 VGPRs per lane (32 re + 32 im).
// k = (reg<<5) | lane. Wire w acts on bit p = 9-w of k:
//   p>=5 -> register bit (pure VALU), p<5 -> lane bit (shfl_xor / cndmask).
// Measurement: probs GEMM vs +/-1 sign matrix via V_WMMA_F32_16X16X4_F32.

#define NQ 10
#define NLAYERS 3
#define WPB 8            // waves (batch elements) per block
#define RSTRIDE 1026     // LDS row stride in floats: 8B-aligned, bank-skewed

typedef __attribute__((ext_vector_type(2))) float v2f;
typedef __attribute__((ext_vector_type(8))) float v8f;

__device__ __forceinline__ float shflx(float v, int m) {
  return __shfl_xor(v, m, 32);
}

// ---- gates on a register bit (wires 0..4), BETA = 4 - wire ----
template<int BETA>
__device__ __forceinline__ void ry_reg(float (&re)[32], float (&im)[32], float c, float s) {
#pragma unroll
  for (int r = 0; r < 32; ++r) {
    if (((r >> BETA) & 1) == 0) {
      const int r1 = r | (1 << BETA);
      float a0 = re[r], a1 = re[r1];
      re[r]  = c * a0 - s * a1;
      re[r1] = s * a0 + c * a1;
      a0 = im[r]; a1 = im[r1];
      im[r]  = c * a0 - s * a1;
      im[r1] = s * a0 + c * a1;
    }
  }
}

template<int BETA>
__device__ __forceinline__ void rz_reg(float (&re)[32], float (&im)[32], float c, float s) {
#pragma unroll
  for (int r = 0; r < 32; ++r) {
    const float t = ((r >> BETA) & 1) ? -s : s;   // e^{-i*th/2} on bit 0, conj on bit 1
    const float nr = c * re[r] + t * im[r];
    const float ni = c * im[r] - t * re[r];
    re[r] = nr; im[r] = ni;
  }
}

// ---- gates on a lane bit (wires 5..9), MASK = 1 << (9 - wire) ----
template<int MASK>
__device__ __forceinline__ void ry_lane(float (&re)[32], float (&im)[32], float c, float s, int lane) {
  const float ss = (lane & MASK) ? s : -s;
#pragma unroll
  for (int r = 0; r < 32; ++r) {
    const float pr = shflx(re[r], MASK);
    const float pi = shflx(im[r], MASK);
    re[r] = c * re[r] + ss * pr;
    im[r] = c * im[r] + ss * pi;
  }
}

template<int MASK>
__device__ __forceinline__ void rz_lane(float (&re)[32], float (&im)[32], float c, float s, int lane) {
  const float t = (lane & MASK) ? -s : s;
#pragma unroll
  for (int r = 0; r < 32; ++r) {
    const float nr = c * re[r] + t * im[r];
    const float ni = c * im[r] - t * re[r];
    re[r] = nr; im[r] = ni;
  }
}

// ---- CNOTs: control bit BC/MC, target bit BT/MT ----
template<int BC, int BT>   // both in register bits: pure register permutation
__device__ __forceinline__ void cnot_rr(float (&re)[32], float (&im)[32]) {
#pragma unroll
  for (int r = 0; r < 32; ++r) {
    if (((r >> BC) & 1) == 1 && ((r >> BT) & 1) == 0) {
      const int r1 = r | (1 << BT);
      float t = re[r]; re[r] = re[r1]; re[r1] = t;
      t = im[r]; im[r] = im[r1]; im[r1] = t;
    }
  }
}

template<int BC, int MT>   // control in reg bits, target in lane bits
__device__ __forceinline__ void cnot_rl(float (&re)[32], float (&im)[32]) {
#pragma unroll
  for (int r = 0; r < 32; ++r) {
    if ((r >> BC) & 1) {
      re[r] = shflx(re[r], MT);
      im[r] = shflx(im[r], MT);
    }
  }
}

template<int MC, int BT>   // control in lane bits, target in reg bits
__device__ __forceinline__ void cnot_lr(float (&re)[32], float (&im)[32], int lane) {
  const bool ctrl = (lane & MC) != 0;
#pragma unroll
  for (int r = 0; r < 32; ++r) {
    if (((r >> BT) & 1) == 0) {
      const int r1 = r | (1 << BT);
      const float a = re[r], b = re[r1];
      re[r]  = ctrl ? b : a;
      re[r1] = ctrl ? a : b;
      const float ai = im[r], bi = im[r1];
      im[r]  = ctrl ? bi : ai;
      im[r1] = ctrl ? ai : bi;
    }
  }
}

template<int MC, int MT>   // both in lane bits
__device__ __forceinline__ void cnot_ll(float (&re)[32], float (&im)[32], int lane) {
  const bool ctrl = (lane & MC) != 0;
#pragma unroll
  for (int r = 0; r < 32; ++r) {
    const float pr = shflx(re[r], MT);
    const float pi = shflx(im[r], MT);
    re[r] = ctrl ? pr : re[r];
    im[r] = ctrl ? pi : im[r];
  }
}

__global__ __launch_bounds__(32 * WPB)
void qsim_kernel(const float* __restrict__ x, const float* __restrict__ wts,
                 float* __restrict__ out, int B) {
  __shared__ float P[WPB * RSTRIDE];
  const int lane = threadIdx.x & 31;
  const int wave = threadIdx.x >> 5;
  int b = blockIdx.x * WPB + wave;
  if (b >= B) b = B - 1;

  float re[32], im[32];
  {
    // Initial layer of per-batch RY gates == rank-1 product state, built directly.
    float ch[NQ], sh[NQ];
#pragma unroll
    for (int i = 0; i < NQ; ++i) {
      const float t = 0.5f * x[b * NQ + i];
      ch[i] = __cosf(t);
      sh[i] = __sinf(t);
    }
    float lo = 1.0f;
#pragma unroll
    for (int wq = 5; wq < NQ; ++wq)
      lo *= ((lane >> (9 - wq)) & 1) ? sh[wq] : ch[wq];
#pragma unroll
    for (int r = 0; r < 32; ++r) {
      float hi = lo;
#pragma unroll
      for (int wq = 0; wq < 5; ++wq)
        hi *= ((r >> (4 - wq)) & 1) ? sh[wq] : ch[wq];   // folds: r, wq constant
      re[r] = hi;
      im[r] = 0.0f;
    }
  }

#pragma unroll
  for (int l = 0; l < NLAYERS; ++l) {
    float yc[NQ], ys[NQ], zc[NQ], zs[NQ];
#pragma unroll
    for (int i = 0; i < NQ; ++i) {
      const float ty = 0.5f * wts[(l * NQ + i) * 2 + 0];
      const float tz = 0.5f * wts[(l * NQ + i) * 2 + 1];
      yc[i] = __cosf(ty); ys[i] = __sinf(ty);
      zc[i] = __cosf(tz); zs[i] = __sinf(tz);
    }
    ry_reg<4>(re, im, yc[0], ys[0]);         rz_reg<4>(re, im, zc[0], zs[0]);
    ry_reg<3>(re, im, yc[1], ys[1]);         rz_reg<3>(re, im, zc[1], zs[1]);
    ry_reg<2>(re, im, yc[2], ys[2]);         rz_reg<2>(re, im, zc[2], zs[2]);
    ry_reg<1>(re, im, yc[3], ys[3]);         rz_reg<1>(re, im, zc[3], zs[3]);
    ry_reg<0>(re, im, yc[4], ys[4]);         rz_reg<0>(re, im, zc[4], zs[4]);
    ry_lane<16>(re, im, yc[5], ys[5], lane); rz_lane<16>(re, im, zc[5], zs[5], lane);
    ry_lane<8>(re, im, yc[6], ys[6], lane);  rz_lane<8>(re, im, zc[6], zs[6], lane);
    ry_lane<4>(re, im, yc[7], ys[7], lane);  rz_lane<4>(re, im, zc[7], zs[7], lane);
    ry_lane<2>(re, im, yc[8], ys[8], lane);  rz_lane<2>(re, im, zc[8], zs[8], lane);
    ry_lane<1>(re, im, yc[9], ys[9], lane);  rz_lane<1>(re, im, zc[9], zs[9], lane);

    cnot_rr<4, 3>(re, im);        // CNOT(0,1)
    cnot_rr<3, 2>(re, im);        // CNOT(1,2)
    cnot_rr<2, 1>(re, im);        // CNOT(2,3)
    cnot_rr<1, 0>(re, im);        // CNOT(3,4)
    cnot_rl<0, 16>(re, im);       // CNOT(4,5)
    cnot_ll<16, 8>(re, im, lane); // CNOT(5,6)
    cnot_ll<8, 4>(re, im, lane);  // CNOT(6,7)
    cnot_ll<4, 2>(re, im, lane);  // CNOT(7,8)
    cnot_ll<2, 1>(re, im, lane);  // CNOT(8,9)
    cnot_lr<1, 4>(re, im, lane);  // CNOT(9,0)
  }

  // probs -> LDS, row per wave/batch element: P[wave][k], k=(r<<5)|lane
  float* prow = &P[wave * RSTRIDE];
#pragma unroll
  for (int r = 0; r < 32; ++r)
    prow[r * 32 + lane] = re[r] * re[r] + im[r] * im[r];
  __syncthreads();

  // Measurement GEMM: Out(8x10) = P(8x1024) x Signs(1024x10), f32 WMMA 16x16x4,
  // 256 K-chunks. Sign matrix computed arithmetically (no memory operand).
  if (wave == 0) {
    const int rowA = lane & 7;              // A rows 8..15 alias 0..7 (discarded)
    const int khalf = (lane >> 4) << 1;     // lanes 0-15: K+0/1, lanes 16-31: K+2/3
    const int n = lane & 15;                // output column (qubit), cols 10..15 = 0-pad
    const int shamt = (n < 10) ? (9 - n) : 0;
    const float valid = (n < 10) ? 1.0f : 0.0f;
    v8f acc = {0.f, 0.f, 0.f, 0.f, 0.f, 0.f, 0.f, 0.f};
#pragma unroll 4
    for (int c = 0; c < 256; ++c) {
      const int k0 = c * 4 + khalf;
      v2f a;
      a.x = P[rowA * RSTRIDE + k0];
      a.y = P[rowA * RSTRIDE + k0 + 1];
      v2f bm;
      bm.x = valid * (1.0f - 2.0f * (float)((k0 >> shamt) & 1));
      bm.y = valid * (1.0f - 2.0f * (float)(((k0 + 1) >> shamt) & 1));
      acc = __builtin_amdgcn_wmma_f32_16x16x4_f32(false, a, false, bm,
                                                  (short)0, acc, false, false);
    }
    // D layout: VGPR j, lanes 0-15 -> M=j, N=lane (lanes 16-31 -> M=j+8, discarded)
    if (lane < 10) {
      const int bbase = blockIdx.x * WPB;
#pragma unroll
      for (int j = 0; j < 8; ++j) {
        const int row = bbase + j;
        if (row < B) out[row * NQ + lane] = acc[j];
      }
    }
  }
}

extern "C" void kernel_launch(void* const* d_in, const int* in_sizes, int n_in,
                              void* d_out, int out_size, void* d_ws, size_t ws_size,
                              hipStream_t stream) {
  const float* x   = (const float*)d_in[0];   // (B, 10) f32
  const float* wts = (const float*)d_in[1];   // (3, 10, 2) f32
  float* out = (float*)d_out;                 // (B, 10) f32
  const int B = in_sizes[0] / NQ;
  const int blocks = (B + WPB - 1) / WPB;
  qsim_kernel<<<blocks, 32 * WPB, 0, stream>>>(x, wts, out, B);
}